// OutProduct_13932873908951
// MI455X (gfx1250) — compile-verified
//
#include <hip/hip_runtime.h>

#define D_NODE 256
#define D_EDGE 128
#define D_HEAD 32
#define N_RES  512
#define NCOL   (D_HEAD * D_EDGE)   // 4096 = j*128 + e
#define LN_EPS 1e-5f

typedef __attribute__((ext_vector_type(2))) float v2f;
typedef __attribute__((ext_vector_type(8))) float v8f;

// ---------------------------------------------------------------------------
// Kernel 1: LayerNorm(node) -> left/right projections.  One 256-thread block
// per residue row. Trivial cost (512 rows, 16 MFLOP total).
// ---------------------------------------------------------------------------
__global__ __launch_bounds__(256) void ln_proj_kernel(
    const float* __restrict__ node, const float* __restrict__ ln_w,
    const float* __restrict__ ln_b,
    const float* __restrict__ w_left,  const float* __restrict__ b_left,
    const float* __restrict__ w_right, const float* __restrict__ b_right,
    float* __restrict__ left, float* __restrict__ right)
{
    __shared__ float xs[D_NODE];
    __shared__ float red[D_NODE];
    const int n = blockIdx.x, t = threadIdx.x;

    float v = node[n * D_NODE + t];
    red[t] = v;
    __syncthreads();
    for (int s = D_NODE / 2; s > 0; s >>= 1) {
        if (t < s) red[t] += red[t + s];
        __syncthreads();
    }
    const float mu = red[0] * (1.0f / D_NODE);
    __syncthreads();

    const float d = v - mu;
    red[t] = d * d;
    __syncthreads();
    for (int s = D_NODE / 2; s > 0; s >>= 1) {
        if (t < s) red[t] += red[t + s];
        __syncthreads();
    }
    const float rs = rsqrtf(red[0] * (1.0f / D_NODE) + LN_EPS);

    xs[t] = d * rs * ln_w[t] + ln_b[t];
    __syncthreads();

    if (t < 2 * D_HEAD) {
        const bool  isL = (t < D_HEAD);
        const float* w  = isL ? w_left  : w_right;
        const float* b  = isL ? b_left  : b_right;
        float*       o  = isL ? left    : right;
        const int    c  = t & (D_HEAD - 1);
        float acc = b[c];
        #pragma unroll 8
        for (int i = 0; i < D_NODE; ++i) acc += xs[i] * w[i * D_HEAD + c];
        o[n * D_HEAD + c] = acc;
    }
}

// ---------------------------------------------------------------------------
// Kernel 2: A[n, j*128+e] = sum_i left[n,i] * w3[i, j*128+e]
// GEMM 512 x 32 x 4096 via v_wmma_f32_16x16x4_f32.
// One wave per 16x16 C tile, 8 chained WMMAs over K=32.
// Grid: 32 n-tiles * 32 groups; each 8-wave block covers 8 column tiles.
// ---------------------------------------------------------------------------
__global__ __launch_bounds__(256) void aproj_kernel(
    const float* __restrict__ left, const float* __restrict__ w_out,
    float* __restrict__ A)
{
    const int wave = threadIdx.x >> 5;
    const int lane = threadIdx.x & 31;
    const int row  = lane & 15;      // M index within tile / N index within tile
    const int hi   = lane >> 4;      // lane group select
    const int koff = 2 * hi;

    const int nt  = blockIdx.x >> 5;            // 0..31 n-tiles
    const int grp = blockIdx.x & 31;            // 0..31
    const int ct  = grp * 8 + wave;             // 0..255 column tiles
    const int n0  = nt * 16;
    const int c0  = ct * 16;

    v8f c = {};
    #pragma unroll
    for (int k0 = 0; k0 < D_HEAD; k0 += 4) {
        v2f a = *(const v2f*)(left + (n0 + row) * D_HEAD + k0 + koff);
        v2f b;
        b.x = w_out[(k0 + koff    ) * NCOL + c0 + row];
        b.y = w_out[(k0 + koff + 1) * NCOL + c0 + row];
        c = __builtin_amdgcn_wmma_f32_16x16x4_f32(
                false, a, false, b, (short)0, c, false, false);
    }
    #pragma unroll
    for (int r = 0; r < 8; ++r)
        A[(n0 + r + 8 * hi) * NCOL + c0 + row] = c[r];
}

// ---------------------------------------------------------------------------
// Kernel 3 (streaming-bound): out[n,m,e] = sum_j right[m,j]*A[n,j,e]
//                                          + b_out[e] + edge[n,m,e]
// Per fixed n: GEMM 512 x 32 x 128. Block = (n, m-tile); its 8 waves are the
// 8 e-tiles, each chaining 8 v_wmma_f32_16x16x4_f32.
// edge/out use non-temporal hints (268 MB streamed once); A stays hot in L2.
// ---------------------------------------------------------------------------
__global__ __launch_bounds__(256) void outer_kernel(
    const float* __restrict__ right, const float* __restrict__ A,
    const float* __restrict__ b_out, const float* __restrict__ edge,
    float* __restrict__ out)
{
    const int wave = threadIdx.x >> 5;
    const int lane = threadIdx.x & 31;
    const int row  = lane & 15;
    const int hi   = lane >> 4;
    const int koff = 2 * hi;

    const int n  = blockIdx.x >> 5;    // 0..511
    const int mt = blockIdx.x & 31;    // 0..31
    const int m0 = mt * 16;
    const int e0 = wave * 16;          // 8 waves -> e in [0,128)

    const float* __restrict__ An = A + (size_t)n * NCOL;

    v8f c = {};
    #pragma unroll
    for (int k0 = 0; k0 < D_HEAD; k0 += 4) {
        v2f a = *(const v2f*)(right + (m0 + row) * D_HEAD + k0 + koff);
        v2f b;
        b.x = An[(k0 + koff    ) * D_EDGE + e0 + row];
        b.y = An[(k0 + koff + 1) * D_EDGE + e0 + row];
        c = __builtin_amdgcn_wmma_f32_16x16x4_f32(
                false, a, false, b, (short)0, c, false, false);
    }

    const int   e  = e0 + row;
    const float bo = b_out[e];
    #pragma unroll
    for (int r = 0; r < 8; ++r) {
        const int    m   = m0 + r + 8 * hi;
        const size_t idx = ((size_t)n * N_RES + m) * D_EDGE + e;
        const float  ev  = __builtin_nontemporal_load(edge + idx);
        __builtin_nontemporal_store(c[r] + bo + ev, out + idx);
    }
}

// ---------------------------------------------------------------------------
extern "C" void kernel_launch(void* const* d_in, const int* in_sizes, int n_in,
                              void* d_out, int out_size, void* d_ws, size_t ws_size,
                              hipStream_t stream)
{
    (void)in_sizes; (void)n_in; (void)out_size; (void)ws_size;

    const float* node    = (const float*)d_in[0];
    const float* edge    = (const float*)d_in[1];
    const float* ln_w    = (const float*)d_in[2];
    const float* ln_b    = (const float*)d_in[3];
    const float* w_left  = (const float*)d_in[4];
    const float* b_left  = (const float*)d_in[5];
    const float* w_right = (const float*)d_in[6];
    const float* b_right = (const float*)d_in[7];
    const float* w_out   = (const float*)d_in[8];
    const float* b_out   = (const float*)d_in[9];
    float* out = (float*)d_out;

    // Workspace: left (64KB) | right (64KB) | A (8MB)
    float* left  = (float*)d_ws;
    float* right = left + (size_t)N_RES * D_HEAD;
    float* A     = right + (size_t)N_RES * D_HEAD;

    ln_proj_kernel<<<N_RES, 256, 0, stream>>>(node, ln_w, ln_b,
                                              w_left, b_left, w_right, b_right,
                                              left, right);
    aproj_kernel<<<32 * 32, 256, 0, stream>>>(left, w_out, A);
    outer_kernel<<<N_RES * 32, 256, 0, stream>>>(right, A, b_out, edge, out);
}